// RadarPillarFE_59390807769312
// MI455X (gfx1250) — compile-verified
//
#include <hip/hip_runtime.h>

// ---------------------------------------------------------------------------
// RadarPillarFE scatter-mean BEV rasterization for MI455X (gfx1250).
//
// Grid: 256x256x1, B=8, N=500000, F=18.
// d_out (B,F,NY,NX) is used directly as the sum accumulator (L2-resident,
// 37.7MB << 192MB L2). Counts (2MB) live in d_ws. Point stream (288MB) is
// moved HBM->LDS with the Tensor Data Mover, double buffered.
// ---------------------------------------------------------------------------

#define F_DIM 18
#define B_DIM 8
#define N_PTS 500000
#define NXG 256
#define NYG 256

constexpr int      TOTAL_PTS   = B_DIM * N_PTS;            // 4,000,000
constexpr int      TILE_PTS    = 256;                      // points per TDM tile
constexpr int      TILE_ELEMS  = TILE_PTS * F_DIM;         // 4608 floats = 18KB
constexpr int      NTILES      = (TOTAL_PTS + TILE_PTS - 1) / TILE_PTS; // 15625
constexpr unsigned TOTAL_ELEMS = (unsigned)TOTAL_PTS * F_DIM;           // 72,000,000
constexpr int      NVOX        = NXG * NYG;                // 65536
constexpr int      OUT_ELEMS   = B_DIM * F_DIM * NVOX;     // 9,437,184
constexpr int      CNT_ELEMS   = B_DIM * NVOX;             // 524,288

typedef unsigned int u32;
typedef u32 u32x4 __attribute__((ext_vector_type(4)));
typedef u32 u32x8 __attribute__((ext_vector_type(8)));

// ---------------------------------------------------------------------------
// Kernel 0: zero the accumulators (d_out sums + d_ws counts). Required per
// call: harness poisons buffers and we accumulate + divide in place.
// ---------------------------------------------------------------------------
__global__ void __launch_bounds__(256) rad_zero_kernel(float4* __restrict__ out4,
                                                       float4* __restrict__ cnt4) {
    int i      = blockIdx.x * blockDim.x + threadIdx.x;
    int stride = gridDim.x * blockDim.x;
    const float4 z = {0.f, 0.f, 0.f, 0.f};
    for (int j = i; j < OUT_ELEMS / 4; j += stride) out4[j] = z;
    for (int j = i; j < CNT_ELEMS / 4; j += stride) cnt4[j] = z;
}

// ---------------------------------------------------------------------------
// TDM issue: build D# groups 0/1 (ISA cdna5 §8.3/§8.4) in SGPRs and launch a
// 1D tile DMA of TILE_ELEMS f32 from global to LDS. tensor_dim0 is set to the
// remaining element count so the final partial tile reads zeros out-of-bounds
// instead of faulting. 2-D descriptor form: VADDR2/VADDR3 omitted.
// ---------------------------------------------------------------------------
__device__ __forceinline__ void tdm_issue_tile(const float* gbase,
                                               unsigned elem_start,
                                               unsigned lds_byte_addr) {
    unsigned long long ga =
        (unsigned long long)(uintptr_t)gbase + (unsigned long long)elem_start * 4ull;
    unsigned dim0 = TOTAL_ELEMS - elem_start;  // remaining; OOB reads return 0

    u32x4 g0;
    g0[0] = 1u;                               // count=1, user descriptor
    g0[1] = lds_byte_addr;                    // lds_addr (bytes, WG-relative)
    g0[2] = (u32)(ga & 0xFFFFFFFFull);        // global_addr[31:0]
    g0[3] = (u32)((ga >> 32) & 0x01FFFFFFull) // global_addr[56:32]
          | (2u << 30);                       // type = 2 ("image")

    u32x8 g1;
    g1[0] = 2u << 16;                         // data_size = 4B; no pad/iter/mask
    g1[1] = (dim0 & 0xFFFFu) << 16;           // tensor_dim0[15:0]
    g1[2] = (dim0 >> 16) | (1u << 16);        // tensor_dim0[31:16], tensor_dim1=1
    g1[3] = ((u32)TILE_ELEMS) << 16;          // tile_dim0 = 4608
    g1[4] = 1u;                               // tile_dim1 = 1, tile_dim2 = 0
    g1[5] = (u32)TILE_ELEMS;                  // tensor_dim0_stride[31:0]
    g1[6] = 0u;                               // stride0[47:32], stride1[15:0]
    g1[7] = 0u;                               // stride1[47:16]

    asm volatile("tensor_load_to_lds %0, %1" :: "s"(g0), "s"(g1) : "memory");
}

// ---------------------------------------------------------------------------
// Kernel 1: TDM-streamed scatter. Each block walks tiles grid-strided; wave 0
// drives the DMA double buffer, all 8 waves consume from LDS and scatter f32
// atomics into the L2-resident accumulators.
// ---------------------------------------------------------------------------
__global__ void __launch_bounds__(256) rad_scatter_kernel(const float* __restrict__ pts,
                                                          float* __restrict__ out,
                                                          float* __restrict__ cnt) {
    __shared__ float lds[2][TILE_ELEMS];

    const int tid = threadIdx.x;
    // Uniform (scalar-branch) wave-0 predicate: TDM ignores EXEC, so exactly
    // one wave per block may execute the tensor op / tensorcnt wait.
    const bool w0 = ((__builtin_amdgcn_readfirstlane((int)tid) >> 5) == 0);

    // Low 32 bits of the generic pointer to LDS == WG-relative LDS byte addr.
    const unsigned lds_base = (unsigned)(uintptr_t)(&lds[0][0]);

    const int first  = blockIdx.x;
    const int stride = gridDim.x;

    if (w0) {
        tdm_issue_tile(pts, (unsigned)first * TILE_ELEMS, lds_base);
        __builtin_amdgcn_s_wait_tensorcnt(0);
    }
    __syncthreads();

    int buf = 0;
    for (int t = first; t < NTILES; t += stride) {
        const int tn = t + stride;
        if (w0 && tn < NTILES) {  // prefetch next tile into the other buffer
            tdm_issue_tile(pts, (unsigned)tn * TILE_ELEMS,
                           lds_base + (unsigned)(buf ^ 1) * (TILE_ELEMS * 4u));
        }

        // ---- consume tile t (overlaps with the in-flight DMA) ----
        const unsigned gp = (unsigned)t * TILE_PTS + (unsigned)tid;
        if (gp < (unsigned)TOTAL_PTS) {
            const float* p = &lds[buf][tid * F_DIM];
            const float x = p[0], y = p[1], z = p[2];
            const bool m = (x >= -51.2f) & (x <= 51.2f) &
                           (y >= -51.2f) & (y <= 51.2f) &
                           (z >= -5.0f)  & (z <= 3.0f);
            if (m) {
                int ix = (int)((x + 51.2f) * 2.5f);
                int iy = (int)((y + 51.2f) * 2.5f);
                ix = ix < 0 ? 0 : (ix > 255 ? 255 : ix);
                iy = iy < 0 ? 0 : (iy > 255 ? 255 : iy);
                const unsigned b   = gp / (unsigned)N_PTS;
                const unsigned vox = (unsigned)iy * NXG + (unsigned)ix;
                float* obase = out + (((b * F_DIM) << 16) | vox);
#pragma unroll
                for (int f = 0; f < F_DIM; ++f)
                    unsafeAtomicAdd(obase + (f << 16), p[f]);
                unsafeAtomicAdd(cnt + ((b << 16) | vox), 1.0f);
            }
        }

        if (w0) __builtin_amdgcn_s_wait_tensorcnt(0);  // next tile landed
        __syncthreads();                                // all consumed buf
        buf ^= 1;
    }
}

// ---------------------------------------------------------------------------
// Kernel 2: in-place mean: out[v] /= max(count[voxel(v)], 1).
// out layout (b,f,y,x); voxel = (b<<16)|yx with yx = v & 0xFFFF.
// ---------------------------------------------------------------------------
__global__ void __launch_bounds__(256) rad_finalize_kernel(float4* __restrict__ out4,
                                                           const float4* __restrict__ cnt4) {
    int i      = blockIdx.x * blockDim.x + threadIdx.x;
    int stride = gridDim.x * blockDim.x;
    for (int q = i; q < OUT_ELEMS / 4; q += stride) {
        const unsigned v  = (unsigned)q * 4u;
        const unsigned yx = v & 0xFFFFu;
        const unsigned b  = (v >> 16) / F_DIM;
        const float4 c = cnt4[((b << 16) | yx) >> 2];
        float4 s = out4[q];
        s.x /= fmaxf(c.x, 1.0f);
        s.y /= fmaxf(c.y, 1.0f);
        s.z /= fmaxf(c.z, 1.0f);
        s.w /= fmaxf(c.w, 1.0f);
        out4[q] = s;
    }
}

// ---------------------------------------------------------------------------
extern "C" void kernel_launch(void* const* d_in, const int* in_sizes, int n_in,
                              void* d_out, int out_size, void* d_ws, size_t ws_size,
                              hipStream_t stream) {
    (void)in_sizes; (void)n_in; (void)out_size; (void)ws_size;

    const float* pts = (const float*)d_in[0];   // (B, N, 18) f32
    float*       out = (float*)d_out;           // (B, 18, 256, 256) f32
    float*       cnt = (float*)d_ws;            // 524,288 f32 counts (2MB)

    rad_zero_kernel    <<<2048, 256, 0, stream>>>((float4*)out, (float4*)cnt);
    rad_scatter_kernel <<<2048, 256, 0, stream>>>(pts, out, cnt);
    rad_finalize_kernel<<<4096, 256, 0, stream>>>((float4*)out, (const float4*)cnt);
}